// GraphComputer_47725676593682
// MI455X (gfx1250) — compile-verified
//
#include <hip/hip_runtime.h>
#include <stdint.h>

// GraphComputer: PBC distances + edge symmetrization + twin-index map.
// Memory-bound gather/scatter (~350 MB @ 23.3 TB/s => ~15us floor).
// CDNA5 paths used: global_load_async_to_lds_b32 (+ s_wait_asynccnt) for the
// streamed edge_index rows, non-temporal load/store hints for all
// streaming traffic, wave32-sized (8-wave) workgroups.

#define NT_ST(p, v) __builtin_nontemporal_store((v), (p))
#define NT_LD(p)    __builtin_nontemporal_load((p))

template <bool UNIFORM_B>
__global__ __launch_bounds__(256) void edge_sym_kernel(
    const float* __restrict__ pos,        // [N,3]
    const float* __restrict__ cell,       // [B,3,3]
    const int*   __restrict__ edge_index, // [2,E] (row-major)
    const int*   __restrict__ cell_off,   // [E,3]
    const int*   __restrict__ num_neigh,  // [B]
    float* __restrict__ out,              // concatenated outputs (float)
    int E, int EG, int B, int bpg)        // bpg = blocks per graph
{
    const int tid = threadIdx.x;
    const int e   = blockIdx.x * 256 + tid;
    const bool active = (e < E);

    // ---- CDNA5 async staging: edge_index rows -> LDS (ASYNCcnt path) ----
    __shared__ int s_ei[2][256];
    if (active) {
        uint32_t lds0 = (uint32_t)(uintptr_t)&s_ei[0][tid];
        uint32_t lds1 = (uint32_t)(uintptr_t)&s_ei[1][tid];
        uint64_t g0   = (uint64_t)(uintptr_t)(edge_index + e);
        uint64_t g1   = (uint64_t)(uintptr_t)(edge_index + E + e);
        asm volatile(
            "global_load_async_to_lds_b32 %0, %1, off\n\t"
            "global_load_async_to_lds_b32 %2, %3, off"
            :
            : "v"(lds0), "v"(g0), "v"(lds1), "v"(g1)
            : "memory");
    }
    asm volatile("s_wait_asynccnt 0" ::: "memory");
    __syncthreads();

    const size_t E2 = 2 * (size_t)E;
    float* __restrict__ o_ei   = out;                 // [2, E2]
    float* __restrict__ o_off  = out + 2 * E2;        // [E2, 3]
    float* __restrict__ o_dist = out + 5 * E2;        // [E2]
    float* __restrict__ o_vec  = out + 6 * E2;        // [E2, 3]
    float* __restrict__ o_nn   = out + 9 * E2;        // [B]
    float* __restrict__ o_swap = o_nn + B;            // [E2]

    // num_neighbors_sym = 2 * num_neighbors
    if (blockIdx.x == 0 && tid < B) {
        NT_ST(o_nn + tid, (float)(2 * num_neigh[tid]));
    }

    if (!active) return;

    // graph id: scalar-uniform when blocks don't straddle graphs
    const int b = UNIFORM_B ? (int)(blockIdx.x / (unsigned)bpg)
                            : (int)((unsigned)e / (unsigned)EG);

    const int i = s_ei[0][tid];
    const int j = s_ei[1][tid];

    const int* __restrict__ co = cell_off + 3 * (size_t)e;
    const int c0 = NT_LD(co + 0);
    const int c1 = NT_LD(co + 1);
    const int c2 = NT_LD(co + 2);
    const float f0 = (float)c0, f1 = (float)c1, f2 = (float)c2;

    const float* __restrict__ cb = cell + 9 * (size_t)b;  // uniform -> SGPR loads
    // offsets_d = sum_c off_c * cell[b][c][d]
    const float ox = f0 * cb[0] + f1 * cb[3] + f2 * cb[6];
    const float oy = f0 * cb[1] + f1 * cb[4] + f2 * cb[7];
    const float oz = f0 * cb[2] + f1 * cb[5] + f2 * cb[8];

    const float* __restrict__ pi = pos + 3 * (size_t)i;
    const float* __restrict__ pj = pos + 3 * (size_t)j;
    const float dx = pi[0] - pj[0] + ox;
    const float dy = pi[1] - pj[1] + oy;
    const float dz = pi[2] - pj[2] + oz;

    const float dist = sqrtf(dx * dx + dy * dy + dz * dz);
    const float inv  = 1.0f / dist;
    const float vx = -dx * inv, vy = -dy * inv, vz = -dz * inv;

    // symmetrized slots: fwd = e + b*EG, bwd = fwd + EG (== twin of fwd)
    const size_t fwd = (size_t)e + (size_t)b * (size_t)EG;
    const size_t bwd = fwd + (size_t)EG;

    // ei_sym
    NT_ST(o_ei + fwd,      (float)i);
    NT_ST(o_ei + bwd,      (float)j);
    NT_ST(o_ei + E2 + fwd, (float)j);
    NT_ST(o_ei + E2 + bwd, (float)i);
    // off_sym (bwd negated via integer negate: avoids -0.0f)
    NT_ST(o_off + 3 * fwd + 0, f0);
    NT_ST(o_off + 3 * fwd + 1, f1);
    NT_ST(o_off + 3 * fwd + 2, f2);
    NT_ST(o_off + 3 * bwd + 0, (float)(-c0));
    NT_ST(o_off + 3 * bwd + 1, (float)(-c1));
    NT_ST(o_off + 3 * bwd + 2, (float)(-c2));
    // dist_sym
    NT_ST(o_dist + fwd, dist);
    NT_ST(o_dist + bwd, dist);
    // vec_sym
    NT_ST(o_vec + 3 * fwd + 0, vx);
    NT_ST(o_vec + 3 * fwd + 1, vy);
    NT_ST(o_vec + 3 * fwd + 2, vz);
    NT_ST(o_vec + 3 * bwd + 0, -vx);
    NT_ST(o_vec + 3 * bwd + 1, -vy);
    NT_ST(o_vec + 3 * bwd + 2, -vz);
    // id_swap_edge_index: twin positions by construction
    NT_ST(o_swap + fwd, (float)bwd);
    NT_ST(o_swap + bwd, (float)fwd);
}

extern "C" void kernel_launch(void* const* d_in, const int* in_sizes, int n_in,
                              void* d_out, int out_size, void* d_ws, size_t ws_size,
                              hipStream_t stream) {
    (void)n_in; (void)out_size; (void)d_ws; (void)ws_size;
    const float* pos        = (const float*)d_in[0];
    const float* cell       = (const float*)d_in[1];
    const int*   edge_index = (const int*)d_in[2];
    const int*   cell_off   = (const int*)d_in[3];
    const int*   num_neigh  = (const int*)d_in[4];

    const int E  = in_sizes[2] / 2;      // edge_index is [2,E]
    const int B  = in_sizes[4];          // num_neighbors is [B]
    const int EG = E / B;

    const int blocks = (E + 255) / 256;
    float* out = (float*)d_out;

    if ((EG & 255) == 0) {
        const int bpg = EG / 256;        // blocks never straddle graphs
        edge_sym_kernel<true><<<blocks, 256, 0, stream>>>(
            pos, cell, edge_index, cell_off, num_neigh, out, E, EG, B, bpg);
    } else {
        edge_sym_kernel<false><<<blocks, 256, 0, stream>>>(
            pos, cell, edge_index, cell_off, num_neigh, out, E, EG, B, 1);
    }
}